// AutoGraphEncoder_71966472012458
// MI455X (gfx1250) — compile-verified
//
#include <hip/hip_runtime.h>
#include <math.h>

// GVP-GNN forward for MI455X (gfx1250, wave32).
// - All large GEMMs (ws_w matmuls, dense1) use V_WMMA_F32_16X16X4_F32, 16x64
//   output tile per wave, fp32 accumulate (matches fp32 reference numerics).
// - The per-edge gather [s[src] | es | s[dst] | vn] is fused into the WMMA
//   A-fragment loads as separate contiguous K-segments with hoisted row base
//   pointers: node state (16 MB) lives in the 192 MB L2, so the indexed reads
//   are cache hits and we avoid materializing the 644 MB message matrix per
//   layer in HBM.
// - gemm_seg is software-pipelined (double-buffered fragments) so next-step
//   A/B loads are in flight while the current 4 WMMAs execute.
// - Scatter-mean via global f32 atomics into a 16 MB (L2-resident) accumulator.
// Workspace requirement: ~1.6 GiB (see layout in kernel_launch).

#define NN 16384
#define EE 262144

typedef float v2f __attribute__((ext_vector_type(2)));
typedef float v8f __attribute__((ext_vector_type(8)));

struct GemmA {
  int mode;            // 0 = plain, 1 = concat2, 2 = message gather
  const float* p0; int ld0; int w0;
  const float* p1; int ld1;
  const int* src; const int* dst;
  const float* es; const float* vn; const float* sn;
};

// Inner K-segment loop, software-pipelined: contiguous A rows starting at
// aBase (already offset by hi*2 for the lane's K-pair), W fragment bases wb[t]
// advanced by kglob. len >= 4 and len % 4 == 0 for every segment we use.
__device__ __forceinline__ void gemm_seg(const float* __restrict__ aBase,
                                         const float* const (&wb)[4],
                                         int kglob, int len, v8f (&acc)[4])
{
  // prologue: fragments for kk = 0
  v2f a; a[0] = aBase[0]; a[1] = aBase[1];
  v2f b[4];
#pragma unroll
  for (int t = 0; t < 4; ++t) {
    const float* wp = wb[t] + kglob;
    b[t][0] = wp[0]; b[t][1] = wp[1];
  }
  // steady state: load kk, compute kk-4
  for (int kk = 4; kk < len; kk += 4) {
    v2f an; an[0] = aBase[kk]; an[1] = aBase[kk + 1];
    v2f bn[4];
#pragma unroll
    for (int t = 0; t < 4; ++t) {
      const float* wp = wb[t] + kglob + kk;
      bn[t][0] = wp[0]; bn[t][1] = wp[1];
    }
#pragma unroll
    for (int t = 0; t < 4; ++t)
      acc[t] = __builtin_amdgcn_wmma_f32_16x16x4_f32(
          false, a, false, b[t], (short)0, acc[t], false, false);
    a = an;
#pragma unroll
    for (int t = 0; t < 4; ++t) b[t] = bn[t];
  }
  // epilogue
#pragma unroll
  for (int t = 0; t < 4; ++t)
    acc[t] = __builtin_amdgcn_wmma_f32_16x16x4_f32(
        false, a, false, b[t], (short)0, acc[t], false, false);
}

// D[M, SO] = A[M, K] @ W[SO, K]^T + bias, optional relu.
// One wave -> 16 rows x 64 cols. grid = (M/64, SO/64), block = 128 (4 waves).
template <int MODE>
__global__ __launch_bounds__(128)
void k_gemm(GemmA A, const float* __restrict__ W, const float* __restrict__ bias,
            float* __restrict__ D, int ldd, int K, int relu)
{
  const int lane = threadIdx.x & 31;
  const int wid  = threadIdx.x >> 5;
  const int row0 = (blockIdx.x * 4 + wid) * 16;
  const int col0 = blockIdx.y * 64;
  const int l15  = lane & 15;
  const int hi   = lane >> 4;             // 0: K pair 0..1 , 1: K pair 2..3
  const int arow = row0 + l15;
  const int koff = hi * 2;

  const float* wb[4];
#pragma unroll
  for (int t = 0; t < 4; ++t)
    wb[t] = W + (size_t)(col0 + t * 16 + l15) * K + koff;

  v8f acc[4] = {};

  if (MODE == 0) {
    const float* a0 = A.p0 + (size_t)arow * A.ld0 + koff;
    gemm_seg(a0, wb, 0, K, acc);
  } else if (MODE == 1) {
    const float* a0 = A.p0 + (size_t)arow * A.ld0 + koff;
    const float* a1 = A.p1 + (size_t)arow * A.ld1 + koff;
    gemm_seg(a0, wb, 0, A.w0, acc);
    gemm_seg(a1, wb, A.w0, K - A.w0, acc);
  } else {
    // [ s[src] (256) | es (64) | s[dst] (256) | vn (68) ]
    const int sr = A.src[arow];
    const int dr = A.dst[arow];
    const float* a0 = A.sn + (size_t)sr * 256 + koff;
    const float* a1 = A.es + (size_t)arow * 64 + koff;
    const float* a2 = A.sn + (size_t)dr * 256 + koff;
    const float* a3 = A.vn + (size_t)arow * 68 + koff;
    gemm_seg(a0, wb, 0, 256, acc);
    gemm_seg(a1, wb, 256, 64, acc);
    gemm_seg(a2, wb, 320, 256, acc);
    gemm_seg(a3, wb, 576, 68, acc);
  }

#pragma unroll
  for (int t = 0; t < 4; ++t) {
    const int n = col0 + t * 16 + l15;
    const float bv = bias ? bias[n] : 0.f;
#pragma unroll
    for (int r = 0; r < 8; ++r) {
      const int m = row0 + hi * 8 + r;
      float v = acc[t][r] + bv;
      if (relu) v = fmaxf(v, 0.f);
      D[(size_t)m * ldd + n] = v;
    }
  }
}

// vh[row,hj,:] = sum_i wh[hj,i] * V(row,i,:) ; vn = ||vh|| (clamped).
// mode 0: vin [M, vi, 3]; mode 1: gather [v[src](32) | ev(4) | v[dst](32)].
__global__ void k_gvp_vec(int mode, const float* __restrict__ vin,
                          const int* __restrict__ src, const int* __restrict__ dst,
                          const float* __restrict__ evv, const float* __restrict__ vnode,
                          const float* __restrict__ wh, int vi, int h,
                          float* __restrict__ vh, float* __restrict__ vn,
                          int vnld, int vnoff, int M)
{
  const long idx = (long)blockIdx.x * blockDim.x + threadIdx.x;
  if (idx >= (long)M * h) return;
  const int row = (int)(idx / h), hj = (int)(idx % h);
  const float* wr = wh + (size_t)hj * vi;
  float a0 = 0.f, a1 = 0.f, a2 = 0.f;
  if (mode == 0) {
    const float* base = vin + (size_t)row * vi * 3;
    for (int i = 0; i < vi; ++i) {
      const float w = wr[i];
      a0 += w * base[i * 3 + 0];
      a1 += w * base[i * 3 + 1];
      a2 += w * base[i * 3 + 2];
    }
  } else {
    const float* bs = vnode + (size_t)src[row] * 96;
    const float* be = evv + (size_t)row * 12;
    const float* bd = vnode + (size_t)dst[row] * 96;
    for (int i = 0; i < 32; ++i) {
      const float w = wr[i];
      a0 += w * bs[i * 3 + 0]; a1 += w * bs[i * 3 + 1]; a2 += w * bs[i * 3 + 2];
    }
    for (int i = 0; i < 4; ++i) {
      const float w = wr[32 + i];
      a0 += w * be[i * 3 + 0]; a1 += w * be[i * 3 + 1]; a2 += w * be[i * 3 + 2];
    }
    for (int i = 0; i < 32; ++i) {
      const float w = wr[36 + i];
      a0 += w * bd[i * 3 + 0]; a1 += w * bd[i * 3 + 1]; a2 += w * bd[i * 3 + 2];
    }
  }
  float* o = vh + ((size_t)row * h + hj) * 3;
  o[0] = a0; o[1] = a1; o[2] = a2;
  const float n2 = fmaxf(a0 * a0 + a1 * a1 + a2 * a2, 1e-8f);
  vn[(size_t)row * vnld + vnoff + hj] = sqrtf(n2);
}

// vout[row,o,:] = sum_j wv[o,j]*vh[row,j,:], optional sigmoid(||.||) gate.
__global__ void k_gvp_wv(const float* __restrict__ vh, const float* __restrict__ wv,
                         float* __restrict__ vout, int h, int vo, int gate, int M)
{
  const long idx = (long)blockIdx.x * blockDim.x + threadIdx.x;
  if (idx >= (long)M * vo) return;
  const int row = (int)(idx / vo), o = (int)(idx % vo);
  const float* wr = wv + (size_t)o * h;
  const float* base = vh + (size_t)row * h * 3;
  float a0 = 0.f, a1 = 0.f, a2 = 0.f;
  for (int j = 0; j < h; ++j) {
    const float w = wr[j];
    a0 += w * base[j * 3 + 0];
    a1 += w * base[j * 3 + 1];
    a2 += w * base[j * 3 + 2];
  }
  if (gate) {
    const float n = sqrtf(fmaxf(a0 * a0 + a1 * a1 + a2 * a2, 1e-8f));
    const float s = 1.f / (1.f + expf(-n));
    a0 *= s; a1 *= s; a2 *= s;
  }
  float* op = vout + ((size_t)row * vo + o) * 3;
  op[0] = a0; op[1] = a1; op[2] = a2;
}

// out = LN(sin + add/max(cnt,1)) * g + b ; safe in-place (one thread per row).
__global__ void k_ln_s(const float* __restrict__ sin, const float* __restrict__ add,
                       const float* __restrict__ cnt, const float* __restrict__ g,
                       const float* __restrict__ b, float* __restrict__ out,
                       int M, int S)
{
  const int r = blockIdx.x * blockDim.x + threadIdx.x;
  if (r >= M) return;
  float inv = 1.f;
  if (cnt) inv = 1.f / fmaxf(cnt[r], 1.f);
  const float* sr = sin + (size_t)r * S;
  const float* ar = add ? add + (size_t)r * S : nullptr;
  float mu = 0.f;
  for (int s = 0; s < S; ++s) { float x = sr[s]; if (ar) x += ar[s] * inv; mu += x; }
  mu /= (float)S;
  float var = 0.f;
  for (int s = 0; s < S; ++s) { float x = sr[s]; if (ar) x += ar[s] * inv; x -= mu; var += x * x; }
  var /= (float)S;
  const float is = rsqrtf(var + 1e-5f);
  float* orow = out + (size_t)r * S;
  for (int s = 0; s < S; ++s) {
    float x = sr[s]; if (ar) x += ar[s] * inv;
    orow[s] = (x - mu) * is * g[s] + b[s];
  }
}

// out = (vin + add/max(cnt,1)) / sqrt(mean_i(clamp(||v_i||^2, eps)))
__global__ void k_ln_v(const float* __restrict__ vin, const float* __restrict__ add,
                       const float* __restrict__ cnt, float* __restrict__ out,
                       int M, int VI)
{
  const int r = blockIdx.x * blockDim.x + threadIdx.x;
  if (r >= M) return;
  float inv = 1.f;
  if (cnt) inv = 1.f / fmaxf(cnt[r], 1.f);
  const float* vr = vin + (size_t)r * VI * 3;
  const float* ar = add ? add + (size_t)r * VI * 3 : nullptr;
  float m = 0.f;
  for (int i = 0; i < VI; ++i) {
    float x0 = vr[i * 3 + 0], x1 = vr[i * 3 + 1], x2 = vr[i * 3 + 2];
    if (ar) { x0 += ar[i * 3 + 0] * inv; x1 += ar[i * 3 + 1] * inv; x2 += ar[i * 3 + 2] * inv; }
    m += fmaxf(x0 * x0 + x1 * x1 + x2 * x2, 1e-8f);
  }
  const float nf = sqrtf(m / (float)VI);
  float* orow = out + (size_t)r * VI * 3;
  for (int i = 0; i < VI; ++i) {
    float x0 = vr[i * 3 + 0], x1 = vr[i * 3 + 1], x2 = vr[i * 3 + 2];
    if (ar) { x0 += ar[i * 3 + 0] * inv; x1 += ar[i * 3 + 1] * inv; x2 += ar[i * 3 + 2] * inv; }
    orow[i * 3 + 0] = x0 / nf; orow[i * 3 + 1] = x1 / nf; orow[i * 3 + 2] = x2 / nf;
  }
}

__global__ void k_fill(float* __restrict__ p, float v, long n)
{
  const long i = (long)blockIdx.x * blockDim.x + threadIdx.x;
  if (i < n) p[i] = v;
}

__global__ void k_count(const int* __restrict__ dsti, float* __restrict__ cnt, int E)
{
  const int e = blockIdx.x * blockDim.x + threadIdx.x;
  if (e < E) atomicAdd(&cnt[dsti[e]], 1.f);
}

__global__ void k_scatter(const float* __restrict__ srcv, const int* __restrict__ dsti,
                          float* __restrict__ acc, int C, int E)
{
  const long idx = (long)blockIdx.x * blockDim.x + threadIdx.x;
  if (idx >= (long)E * C) return;
  const int e = (int)(idx / C), c = (int)(idx % C);
  atomicAdd(&acc[(size_t)dsti[e] * C + c], srcv[idx]);
}

// dense2 + log_softmax + NLL loss. out[0] = loss (atomic), out+1 = logits.
__global__ void k_head(const float* __restrict__ H, const float* __restrict__ W2,
                       const float* __restrict__ b2, const int* __restrict__ labels,
                       float* __restrict__ out, int M)
{
  const int r = blockIdx.x * blockDim.x + threadIdx.x;
  if (r >= M) return;
  const float* hr = H + (size_t)r * 512;
  float lg[20];
  for (int c = 0; c < 20; ++c) {
    const float* wr = W2 + (size_t)c * 512;
    float a = b2[c];
    for (int k = 0; k < 512; ++k) a += hr[k] * wr[k];
    lg[c] = a;
  }
  float mx = lg[0];
  for (int c = 1; c < 20; ++c) mx = fmaxf(mx, lg[c]);
  float se = 0.f;
  for (int c = 0; c < 20; ++c) se += expf(lg[c] - mx);
  const float lse = mx + logf(se);
  for (int c = 0; c < 20; ++c) out[1 + (size_t)r * 20 + c] = lg[c];
  const float lp = lg[labels[r]] - lse;
  atomicAdd(out, -lp / (float)M);
}

static inline GemmA plainA(const float* p, int ld) {
  GemmA a = {}; a.mode = 0; a.p0 = p; a.ld0 = ld; return a;
}
static inline GemmA cat2A(const float* p0, int ld0, int w0, const float* p1, int ld1) {
  GemmA a = {}; a.mode = 1; a.p0 = p0; a.ld0 = ld0; a.w0 = w0; a.p1 = p1; a.ld1 = ld1; return a;
}
static inline GemmA msgA(const int* s, const int* d, const float* es,
                         const float* vn, const float* sn) {
  GemmA a = {}; a.mode = 2; a.src = s; a.dst = d; a.es = es; a.vn = vn; a.sn = sn; return a;
}

static inline int nb(long n) { return (int)((n + 255) / 256); }

extern "C" void kernel_launch(void* const* d_in, const int* in_sizes, int n_in,
                              void* d_out, int out_size, void* d_ws, size_t ws_size,
                              hipStream_t stream)
{
  (void)in_sizes; (void)n_in; (void)out_size; (void)ws_size;
  const float* node_s = (const float*)d_in[0];   // [N,20]
  const float* node_v = (const float*)d_in[1];   // [N,3,3]
  const int*   ei     = (const int*)d_in[2];     // [2,E]
  const float* edge_s = (const float*)d_in[3];   // [E,32]
  const float* edge_v = (const float*)d_in[4];   // [E,1,3]
  const int*   labels = (const int*)d_in[5];     // [N]
  const float* P      = (const float*)d_in[6];   // params, flat, construction order
  float* out = (float*)d_out;

  const int* src = ei;
  const int* dst = ei + EE;

  // ---- parameter unpacking (insertion/construction order of init_params) ----
  size_t po = 0;
  auto take = [&](size_t n) { const float* r = P + po; po += n; return r; };
  const float* wvln_g = take(20);  const float* wvln_b = take(20);
  const float* wv_wh = take(32 * 3);   const float* wv_ws = take(256 * 52);
  const float* wv_wv = take(32 * 32);  const float* wv_b  = take(256);
  const float* weln_g = take(32);  const float* weln_b = take(32);
  const float* we_wh = take(4 * 1);    const float* we_ws = take(64 * 36);
  const float* we_wv = take(4 * 4);    const float* we_b  = take(64);
  const float* woln_g = take(256); const float* woln_b = take(256);
  const float* wo_wh = take(32 * 32);  const float* wo_ws = take(256 * 288);
  const float* wo_b  = take(256);
  const float* d1_w = take(512 * 256); const float* d1_b = take(512);
  const float* d2_w = take(20 * 512);  const float* d2_b = take(20);
  struct Layer {
    const float *m0wh, *m0ws, *m0wv, *m0b;
    const float *m1wh, *m1ws, *m1wv, *m1b;
    const float *m2wh, *m2ws, *m2wv, *m2b;
    const float *n0g, *n0b;
    const float *f0wh, *f0ws, *f0wv, *f0b;
    const float *f1wh, *f1ws, *f1wv, *f1b;
    const float *n1g, *n1b;
  } L[4];
  for (int l = 0; l < 4; ++l) {
    L[l].m0wh = take(68 * 68); L[l].m0ws = take(256 * 644); L[l].m0wv = take(32 * 68); L[l].m0b = take(256);
    L[l].m1wh = take(32 * 32); L[l].m1ws = take(256 * 288); L[l].m1wv = take(32 * 32); L[l].m1b = take(256);
    L[l].m2wh = take(32 * 32); L[l].m2ws = take(256 * 288); L[l].m2wv = take(32 * 32); L[l].m2b = take(256);
    L[l].n0g = take(256); L[l].n0b = take(256);
    L[l].f0wh = take(64 * 32); L[l].f0ws = take(1024 * 320); L[l].f0wv = take(64 * 64); L[l].f0b = take(1024);
    L[l].f1wh = take(64 * 64); L[l].f1ws = take(256 * 1088); L[l].f1wv = take(32 * 64); L[l].f1b = take(256);
    L[l].n1g = take(256); L[l].n1b = take(256);
  }

  // ---- workspace layout (floats); total ~1.57 GiB ----
  float* ws = (float*)d_ws;
  size_t wo = 0;
  auto alloc = [&](size_t n) { float* r = ws + wo; wo += n; return r; };
  float* s_node = alloc((size_t)NN * 256);
  float* v_node = alloc((size_t)NN * 96);
  float* es_e   = alloc((size_t)EE * 64);
  float* ev_e   = alloc((size_t)EE * 12);
  float* dhs    = alloc((size_t)NN * 256);
  float* dhv    = alloc((size_t)NN * 96);
  float* cntb   = alloc((size_t)NN);
  float* vn0    = alloc((size_t)EE * 68);
  float* XB     = alloc((size_t)EE * 288);
  float* XC     = alloc((size_t)EE * 288);
  float* VHe    = alloc((size_t)EE * 204);
  float* V1     = alloc((size_t)EE * 96);
  float* V2     = alloc((size_t)EE * 96);
  float* SE     = alloc((size_t)EE * 256);
  float* SL     = alloc((size_t)NN * 20);
  float* VL     = alloc((size_t)NN * 9);
  float* VNn    = alloc((size_t)NN * 64);
  float* vhn    = alloc((size_t)NN * 192);
  float* vfn    = alloc((size_t)NN * 192);
  float* FS     = alloc((size_t)NN * 256);
  float* FV     = alloc((size_t)NN * 96);
  float* SL2    = alloc((size_t)NN * 256);
  float* VL2    = alloc((size_t)NN * 96);
  float* XN1    = alloc((size_t)NN * 1088);
  float* H1     = alloc((size_t)NN * 512);
  float* OUTS   = alloc((size_t)NN * 256);
  // edge-embedding temporaries aliased into XB (free at that point)
  float* ESL = XB;
  float* EVL = XB + (size_t)EE * 32;
  float* EVH = XB + (size_t)EE * 35;
  float* EVN = XB + (size_t)EE * 47;

  auto gemm = [&](GemmA A, const float* W, const float* b, float* D, int ldd,
                  int M, int K, int SO, int relu) {
    dim3 g((unsigned)(M / 64), (unsigned)(SO / 64));
    if (A.mode == 0)      k_gemm<0><<<g, 128, 0, stream>>>(A, W, b, D, ldd, K, relu);
    else if (A.mode == 1) k_gemm<1><<<g, 128, 0, stream>>>(A, W, b, D, ldd, K, relu);
    else                  k_gemm<2><<<g, 128, 0, stream>>>(A, W, b, D, ldd, K, relu);
  };

  // ================= node embedding =================
  k_ln_s<<<nb(NN), 256, 0, stream>>>(node_s, nullptr, nullptr, wvln_g, wvln_b, SL, NN, 20);
  k_ln_v<<<nb(NN), 256, 0, stream>>>(node_v, nullptr, nullptr, VL, NN, 3);
  k_gvp_vec<<<nb((long)NN * 32), 256, 0, stream>>>(0, VL, nullptr, nullptr, nullptr, nullptr,
                                                   wv_wh, 3, 32, vhn, VNn, 64, 0, NN);
  gemm(cat2A(SL, 20, 20, VNn, 64), wv_ws, wv_b, s_node, 256, NN, 52, 256, 0);
  k_gvp_wv<<<nb((long)NN * 32), 256, 0, stream>>>(vhn, wv_wv, v_node, 32, 32, 0, NN);

  // ================= edge embedding =================
  k_ln_s<<<nb(EE), 256, 0, stream>>>(edge_s, nullptr, nullptr, weln_g, weln_b, ESL, EE, 32);
  k_ln_v<<<nb(EE), 256, 0, stream>>>(edge_v, nullptr, nullptr, EVL, EE, 1);
  k_gvp_vec<<<nb((long)EE * 4), 256, 0, stream>>>(0, EVL, nullptr, nullptr, nullptr, nullptr,
                                                  we_wh, 1, 4, EVH, EVN, 4, 0, EE);
  gemm(cat2A(ESL, 32, 32, EVN, 4), we_ws, we_b, es_e, 64, EE, 36, 64, 0);
  k_gvp_wv<<<nb((long)EE * 4), 256, 0, stream>>>(EVH, we_wv, ev_e, 4, 4, 0, EE);

  // ================= message-passing layers =================
  for (int l = 0; l < 4; ++l) {
    // --- msg GVP 0 (gathered input, relu + vector gate) ---
    k_gvp_vec<<<nb((long)EE * 68), 256, 0, stream>>>(1, nullptr, src, dst, ev_e, v_node,
                                                     L[l].m0wh, 68, 68, VHe, vn0, 68, 0, EE);
    gemm(msgA(src, dst, es_e, vn0, s_node), L[l].m0ws, L[l].m0b, XB, 288, EE, 644, 256, 1);
    k_gvp_wv<<<nb((long)EE * 32), 256, 0, stream>>>(VHe, L[l].m0wv, V1, 68, 32, 1, EE);
    // --- msg GVP 1 ---
    k_gvp_vec<<<nb((long)EE * 32), 256, 0, stream>>>(0, V1, nullptr, nullptr, nullptr, nullptr,
                                                     L[l].m1wh, 32, 32, VHe, XB, 288, 256, EE);
    gemm(plainA(XB, 288), L[l].m1ws, L[l].m1b, XC, 288, EE, 288, 256, 1);
    k_gvp_wv<<<nb((long)EE * 32), 256, 0, stream>>>(VHe, L[l].m1wv, V2, 32, 32, 1, EE);
    // --- msg GVP 2 (no activations) ---
    k_gvp_vec<<<nb((long)EE * 32), 256, 0, stream>>>(0, V2, nullptr, nullptr, nullptr, nullptr,
                                                     L[l].m2wh, 32, 32, VHe, XC, 288, 256, EE);
    gemm(plainA(XC, 288), L[l].m2ws, L[l].m2b, SE, 256, EE, 288, 256, 0);
    k_gvp_wv<<<nb((long)EE * 32), 256, 0, stream>>>(VHe, L[l].m2wv, V1, 32, 32, 0, EE);
    // --- scatter-mean into nodes ---
    k_fill<<<nb((long)NN * 256), 256, 0, stream>>>(dhs, 0.f, (long)NN * 256);
    k_fill<<<nb((long)NN * 96), 256, 0, stream>>>(dhv, 0.f, (long)NN * 96);
    k_fill<<<nb(NN), 256, 0, stream>>>(cntb, 0.f, NN);
    k_count<<<nb(EE), 256, 0, stream>>>(dst, cntb, EE);
    k_scatter<<<nb((long)EE * 256), 256, 0, stream>>>(SE, dst, dhs, 256, EE);
    k_scatter<<<nb((long)EE * 96), 256, 0, stream>>>(V1, dst, dhv, 96, EE);
    k_ln_s<<<nb(NN), 256, 0, stream>>>(s_node, dhs, cntb, L[l].n0g, L[l].n0b, s_node, NN, 256);
    k_ln_v<<<nb(NN), 256, 0, stream>>>(v_node, dhv, cntb, v_node, NN, 32);
    // --- feedforward GVP 0 (relu + gate) ---
    k_gvp_vec<<<nb((long)NN * 64), 256, 0, stream>>>(0, v_node, nullptr, nullptr, nullptr, nullptr,
                                                     L[l].f0wh, 32, 64, vhn, VNn, 64, 0, NN);
    gemm(cat2A(s_node, 256, 256, VNn, 64), L[l].f0ws, L[l].f0b, XN1, 1088, NN, 320, 1024, 1);
    k_gvp_wv<<<nb((long)NN * 64), 256, 0, stream>>>(vhn, L[l].f0wv, vfn, 64, 64, 1, NN);
    // --- feedforward GVP 1 (no activations) ---
    k_gvp_vec<<<nb((long)NN * 64), 256, 0, stream>>>(0, vfn, nullptr, nullptr, nullptr, nullptr,
                                                     L[l].f1wh, 64, 64, vhn, XN1, 1088, 1024, NN);
    gemm(plainA(XN1, 1088), L[l].f1ws, L[l].f1b, FS, 256, NN, 1088, 256, 0);
    k_gvp_wv<<<nb((long)NN * 32), 256, 0, stream>>>(vhn, L[l].f1wv, FV, 64, 32, 0, NN);
    k_ln_s<<<nb(NN), 256, 0, stream>>>(s_node, FS, nullptr, L[l].n1g, L[l].n1b, s_node, NN, 256);
    k_ln_v<<<nb(NN), 256, 0, stream>>>(v_node, FV, nullptr, v_node, NN, 32);
  }

  // ================= output head =================
  k_ln_s<<<nb(NN), 256, 0, stream>>>(s_node, nullptr, nullptr, woln_g, woln_b, SL2, NN, 256);
  k_ln_v<<<nb(NN), 256, 0, stream>>>(v_node, nullptr, nullptr, VL2, NN, 32);
  k_gvp_vec<<<nb((long)NN * 32), 256, 0, stream>>>(0, VL2, nullptr, nullptr, nullptr, nullptr,
                                                   wo_wh, 32, 32, vhn, VNn, 64, 0, NN);
  gemm(cat2A(SL2, 256, 256, VNn, 64), wo_ws, wo_b, OUTS, 256, NN, 288, 256, 1);
  gemm(plainA(OUTS, 256), d1_w, d1_b, H1, 512, NN, 256, 512, 1);
  k_fill<<<1, 256, 0, stream>>>(out, 0.f, 1);
  k_head<<<nb(NN), 256, 0, stream>>>(H1, d2_w, d2_b, labels, out, NN);
}